// bbs_46866683134132
// MI455X (gfx1250) — compile-verified
//
#include <hip/hip_runtime.h>
#include <hip/hip_bf16.h>

typedef __attribute__((ext_vector_type(2))) float v2f;
typedef __attribute__((ext_vector_type(8))) float v8f;
typedef int v4i __attribute__((vector_size(4 * sizeof(int))));

#define B_    16
#define CB    24
#define CC    276          // CB*(CB-1)/2
#define THW   4096         // 4*32*32
#define EPSD  1e-6f
#define MACRO 32
#define NTILE 9            // ceil(288/32) macro tiles per dim (pad 276 -> 288)
#define NPAIR 45           // NTILE*(NTILE+1)/2 upper-tri macro tile pairs
#define NOFF  37950.0f     // 276*275/2 off-diagonal count
#define KC    64           // K chunk per LDS stage
#define LSTR  68           // LDS row stride (floats): 16B aligned, conflict-free

#if __has_builtin(__builtin_amdgcn_global_load_async_to_lds_b128)
#define USE_ASYNC 1
#else
#define USE_ASYNC 0
#endif

#if USE_ASYNC
typedef __attribute__((address_space(1))) v4i* as1_v4i_p;
typedef __attribute__((address_space(3))) v4i* as3_v4i_p;

__device__ __forceinline__ void async_b128(const float* g, float* l) {
    __builtin_amdgcn_global_load_async_to_lds_b128(
        (as1_v4i_p)g, (as3_v4i_p)l, 0, 0);
}
__device__ __forceinline__ void wait_async0() {
#if __has_builtin(__builtin_amdgcn_s_wait_asynccnt)
    __builtin_amdgcn_s_wait_asynccnt(0);
#else
    asm volatile("s_wait_asynccnt 0x0" ::: "memory");
#endif
}
#endif

__device__ __forceinline__ void pair_from_c(int c, int& i, int& j) {
    int rem = c, cnt = CB - 1, ii = 0;
    while (rem >= cnt) { rem -= cnt; cnt--; ii++; }
    i = ii; j = ii + 1 + rem;
}

// ---- Kernel 1: per-channel sum of squares of the ratio over (B, THW) ----
__global__ __launch_bounds__(256) void k_ssq(const float* __restrict__ x,
                                             float* __restrict__ ssq) {
    const int c = blockIdx.x, b = blockIdx.y;
    int i1, i2; pair_from_c(c, i1, i2);            // uniform per block (scalar loop)
    const float4* xi = (const float4*)(x + ((size_t)b * CB + i1) * THW);
    const float4* xj = (const float4*)(x + ((size_t)b * CB + i2) * THW);
    float s = 0.f;
    for (int p = threadIdx.x; p < THW / 4; p += 256) {
        float4 a = xi[p], d = xj[p];
        float r0 = a.x / (d.x + EPSD);
        float r1 = a.y / (d.y + EPSD);
        float r2 = a.z / (d.z + EPSD);
        float r3 = a.w / (d.w + EPSD);
        s += r0 * r0 + r1 * r1 + r2 * r2 + r3 * r3;
    }
    __shared__ float sm[256];
    sm[threadIdx.x] = s;
    __syncthreads();
    for (int off = 128; off > 0; off >>= 1) {
        if ((int)threadIdx.x < off) sm[threadIdx.x] += sm[threadIdx.x + off];
        __syncthreads();
    }
    if (threadIdx.x == 0) atomicAdd(&ssq[c], sm[0]);
}

// ---- Kernel 2: recompute ratio, L2-normalize per channel, write out_n ----
__global__ __launch_bounds__(256) void k_norm(const float* __restrict__ x,
                                              const float* __restrict__ ssq,
                                              float* __restrict__ out) {
    const int c = blockIdx.y, b = blockIdx.z;
    const int p = blockIdx.x * 256 + threadIdx.x;   // float4 index
    int i1, i2; pair_from_c(c, i1, i2);
    const float inv = 1.0f / fmaxf(sqrtf(ssq[c]), EPSD);
    const float4 a = ((const float4*)(x + ((size_t)b * CB + i1) * THW))[p];
    const float4 d = ((const float4*)(x + ((size_t)b * CB + i2) * THW))[p];
    float4 o;
    o.x = (a.x / (d.x + EPSD)) * inv;
    o.y = (a.y / (d.y + EPSD)) * inv;
    o.z = (a.z / (d.z + EPSD)) * inv;
    o.w = (a.w / (d.w + EPSD)) * inv;
    ((float4*)(out + ((size_t)b * CC + c) * THW))[p] = o;
}

// ---- Kernel 3: batched SYRK upper-tri |cov| sum via WMMA f32 16x16x4 ----
// 4 waves per block, 32x32 macro tile, double-buffered LDS fed by async
// global->LDS (ASYNCcnt) when available, K chunks of 64.
__global__ __launch_bounds__(128) void k_syrk(const float* __restrict__ fn,
                                              float* __restrict__ acc) {
    const int b = blockIdx.y;
    int p = blockIdx.x;
    int tr = 0;
    while (p >= NTILE - tr) { p -= NTILE - tr; tr++; }   // uniform scalar decode
    const int tc = tr + p;                               // tr <= tc

    const float* base = fn + (size_t)b * CC * THW;

    __shared__ float lA[2][MACRO][LSTR];
    __shared__ float lB[2][MACRO][LSTR];

    const int t    = threadIdx.x;
    const int lane = t & 31;
    const int wave = t >> 5;
    const int wi   = wave >> 1;          // sub-tile row (0/1)
    const int wj   = wave & 1;           // sub-tile col (0/1)

    // staging: each thread loads 16 floats (4x b128) per operand per chunk.
    // Padded rows (>=276) are CLAMPED to row 275: they only feed masked outputs.
    const int lrow = t >> 2;             // 0..31
    const int lcol = (t & 3) * 16;       // 0,16,32,48
    const int gi = min(tr * MACRO + lrow, CC - 1);
    const int gj = min(tc * MACRO + lrow, CC - 1);
    const float* gArow = base + (size_t)gi * THW;
    const float* gBrow = base + (size_t)gj * THW;

    // fragment gather coordinates (A 16x4 layout; B mirrored)
    const int mrow = wi * 16 + (lane & 15);
    const int nrow = wj * 16 + (lane & 15);
    const int koff = (lane & 16) ? 2 : 0;

    v8f accv = {0.f, 0.f, 0.f, 0.f, 0.f, 0.f, 0.f, 0.f};

    auto stage = [&](int buf, int k0) {
#if USE_ASYNC
        #pragma unroll
        for (int q = 0; q < 4; ++q) {
            async_b128(gArow + k0 + lcol + 4 * q, &lA[buf][lrow][lcol + 4 * q]);
            async_b128(gBrow + k0 + lcol + 4 * q, &lB[buf][lrow][lcol + 4 * q]);
        }
#else
        #pragma unroll
        for (int q = 0; q < 4; ++q) {
            *(float4*)&lA[buf][lrow][lcol + 4 * q] =
                *(const float4*)(gArow + k0 + lcol + 4 * q);
            *(float4*)&lB[buf][lrow][lcol + 4 * q] =
                *(const float4*)(gBrow + k0 + lcol + 4 * q);
        }
#endif
    };

    stage(0, 0);
#if USE_ASYNC
    wait_async0();
#endif
    __syncthreads();

    int buf = 0;
    for (int k0 = 0; k0 < THW; k0 += KC) {
        if (k0 + KC < THW) stage(buf ^ 1, k0 + KC);   // prefetch next chunk
        #pragma unroll
        for (int kk = 0; kk < KC; kk += 4) {
            const float* ap = &lA[buf][mrow][kk + koff];
            const float* bp = &lB[buf][nrow][kk + koff];
            v2f a, bb;
            a.x  = ap[0]; a.y  = ap[1];
            bb.x = bp[0]; bb.y = bp[1];
            accv = __builtin_amdgcn_wmma_f32_16x16x4_f32(
                false, a, false, bb, (short)0, accv, false, false);
        }
#if USE_ASYNC
        wait_async0();
#endif
        __syncthreads();
        buf ^= 1;
    }

    // epilogue: strict upper-triangle |.| sum (raw dot products; scale later)
    float part = 0.f;
    const int jglob = tc * MACRO + wj * 16 + (lane & 15);
    #pragma unroll
    for (int r = 0; r < 8; ++r) {
        const int iglob = tr * MACRO + wi * 16 + r + ((lane & 16) ? 8 : 0);
        if (iglob < CC && jglob < CC && iglob < jglob) part += fabsf(accv[r]);
    }
    for (int off = 16; off > 0; off >>= 1) part += __shfl_down(part, off, 32);
    if (lane == 0) atomicAdd(&acc[b], part);
}

// ---- Kernel 4: finalize scalar loss ----
__global__ void k_loss(const float* __restrict__ acc, float* __restrict__ outloss) {
    if (threadIdx.x == 0 && blockIdx.x == 0) {
        float s = 0.f;
        for (int b = 0; b < B_; ++b) {
            float cov_sum = acc[b] * (1.0f / (float)(THW - 1));   // /4095
            s += fmaxf(cov_sum / NOFF, 0.f);
        }
        outloss[0] = s / (float)B_;
    }
}

extern "C" void kernel_launch(void* const* d_in, const int* in_sizes, int n_in,
                              void* d_out, int out_size, void* d_ws, size_t ws_size,
                              hipStream_t stream) {
    const float* x = (const float*)d_in[0];
    float* out = (float*)d_out;
    float* ws  = (float*)d_ws;
    float* ssq = ws;         // 276 floats (padded to 288)
    float* acc = ws + 288;   // 16 floats

    size_t zbytes = (288 + 16) * sizeof(float);
    if (zbytes > ws_size) zbytes = ws_size;
    (void)hipMemsetAsync(d_ws, 0, zbytes, stream);

    k_ssq <<<dim3(CC, B_),                  256, 0, stream>>>(x, ssq);
    k_norm<<<dim3(THW / (256 * 4), CC, B_), 256, 0, stream>>>(x, ssq, out);
    k_syrk<<<dim3(NPAIR, B_),               128, 0, stream>>>(out, acc);
    k_loss<<<1, 1, 0, stream>>>(acc, out + (size_t)out_size - 1);
}